// SEIR_SDE_Model_1262720385729
// MI455X (gfx1250) — compile-verified
//
#include <hip/hip_runtime.h>
#include <hip/hip_bf16.h>

#ifndef __has_builtin
#define __has_builtin(x) 0
#endif

#if __has_builtin(__builtin_amdgcn_global_load_async_to_lds_b128)
#define HAS_ASYNC 1
#else
#define HAS_ASYNC 0
#endif
#if __has_builtin(__builtin_amdgcn_s_wait_asynccnt)
#define HAS_WAITASYNC 1
#else
#define HAS_WAITASYNC 0
#endif

#define TPB    128
#define BTRAJ  32768
#define STEPS  1000
#define CHUNK  8
#define NBUF   4
#define NCHUNK (STEPS / CHUNK)   // 125

typedef __attribute__((ext_vector_type(4))) float v4f;
typedef __attribute__((ext_vector_type(4))) int   v4i;
// Address-space-qualified pointee types for the async builtin (sig: vV4i*1 V4i*3 Ii Ii)
typedef __attribute__((address_space(1))) v4i glob_v4i;
typedef __attribute__((address_space(3))) v4i lds_v4i;

__device__ __forceinline__ float fast_sqrtf(float x) {
#if __has_builtin(__builtin_amdgcn_sqrtf)
  return __builtin_amdgcn_sqrtf(x);   // raw v_sqrt_f32
#else
  return sqrtf(x);
#endif
}

// Async 16B global -> LDS copy (per-lane), tracked by ASYNCcnt.
__device__ __forceinline__ void async_cp16(const v4f* g, v4f* l) {
#if HAS_ASYNC
  __builtin_amdgcn_global_load_async_to_lds_b128(
      (glob_v4i*)g, (lds_v4i*)l, /*imm offset*/ 0, /*cpol*/ 0);
#else
  (void)g; (void)l;
#endif
}

template <int N>
__device__ __forceinline__ void wait_async() {
#if HAS_ASYNC
#if HAS_WAITASYNC
  __builtin_amdgcn_s_wait_asynccnt(N);
#else
  asm volatile("s_wait_asynccnt %0" :: "i"(N) : "memory");
#endif
  asm volatile("" ::: "memory");   // keep LDS reads below the wait
#endif
}

__global__ __launch_bounds__(TPB) void seir_sde_kernel(
    const v4f* __restrict__ init, const v4f* __restrict__ noise,
    v4f* __restrict__ out) {
  const int tid = threadIdx.x;
  const int gid = blockIdx.x * TPB + tid;

  v4f st = init[gid];
  float S = st.x, E = st.y, I = st.z, R = st.w;

  const v4f* np = noise + gid;   // stride BTRAJ v4f per step
  v4f*       op = out   + gid;

  const float SQRT_DT = 0.31622776601683794f;  // f32(sqrt(0.1))

  auto step = [&](v4f w) {
    float dW0 = w.x * SQRT_DT;
    float dW1 = w.y * SQRT_DT;
    float dW2 = w.z * SQRT_DT;
    float inf = (0.3f * (1.0f / 50.0f)) * S * I;  // BETA*S*I/NPOP
    float ex  = 0.1f * E;                          // SIGMA*E
    float rc  = 0.1f * I;                          // GAMMA*I
    float n0  = fast_sqrtf(fmaxf(inf, 1e-8f)) * dW0;
    float n1  = fast_sqrtf(fmaxf(ex , 1e-8f)) * dW1;
    float n2  = fast_sqrtf(fmaxf(rc , 1e-8f)) * dW2;
    S = S - 0.1f * inf       - n0;
    E = E + 0.1f * (inf - ex) + (n0 - n1);
    I = I + 0.1f * (ex - rc)  + (n1 - n2);
    R = R + 0.1f * rc         + n2;
  };

#if HAS_ASYNC
  // Ring buffer: NBUF chunks of CHUNK steps, 16B per thread per step. 64 KB LDS.
  __shared__ v4f sbuf[NBUF * CHUNK * TPB];
  v4f* lb = &sbuf[tid];

  auto issue = [&](int c) {
    const v4f* g = np + (size_t)c * (CHUNK * BTRAJ);
    v4f*       l = lb + (c & (NBUF - 1)) * (CHUNK * TPB);
#pragma unroll
    for (int s = 0; s < CHUNK; ++s)
      async_cp16(g + (size_t)s * BTRAJ, l + s * TPB);
  };

  auto consume = [&](int c) {
#pragma unroll
    for (int s = 0; s < CHUNK; ++s) {
      v4f w = lb[((c & (NBUF - 1)) * CHUNK + s) * TPB];  // ds_load_b128
      step(w);
      v4f o; o.x = S; o.y = E; o.z = I; o.w = R;
      __builtin_nontemporal_store(o, op + (size_t)(c * CHUNK + s) * BTRAJ);
    }
  };

  issue(0); issue(1); issue(2);                 // prologue: 3 chunks ahead
  for (int c = 0; c < NCHUNK - (NBUF - 1); ++c) {
    issue(c + NBUF - 1);                        // keep 3 chunks in flight
    wait_async<(NBUF - 1) * CHUNK>();           // <=24: chunk c landed in LDS
    consume(c);
  }
  wait_async<(NBUF - 2) * CHUNK>(); consume(NCHUNK - 3);
  wait_async<(NBUF - 3) * CHUNK>(); consume(NCHUNK - 2);
  wait_async<0>();                  consume(NCHUNK - 1);
#else
  // Fallback: prefetch-pipelined direct streaming loads.
  for (int t = 0; t < STEPS; ++t) {
    if (t + 16 < STEPS)
      __builtin_prefetch(np + (size_t)(t + 16) * BTRAJ, 0, 0);
    v4f w = __builtin_nontemporal_load(np + (size_t)t * BTRAJ);
    step(w);
    v4f o; o.x = S; o.y = E; o.z = I; o.w = R;
    __builtin_nontemporal_store(o, op + (size_t)t * BTRAJ);
  }
#endif
}

extern "C" void kernel_launch(void* const* d_in, const int* in_sizes, int n_in,
                              void* d_out, int out_size, void* d_ws, size_t ws_size,
                              hipStream_t stream) {
  (void)in_sizes; (void)n_in; (void)out_size; (void)d_ws; (void)ws_size;
  const v4f* init  = (const v4f*)d_in[0];   // (32768, 4) f32
  const v4f* noise = (const v4f*)d_in[1];   // (1000, 32768, 4) f32
  v4f*       out   = (v4f*)d_out;           // (1000, 32768, 4) f32
  seir_sde_kernel<<<BTRAJ / TPB, TPB, 0, stream>>>(init, noise, out);
}